// DTW_SPRING_ROW_38448547233960
// MI455X (gfx1250) — compile-verified
//
#include <hip/hip_runtime.h>
#include <stdint.h>

// ---------------- problem constants ----------------
#define CHUNK    16384            // elements per workgroup chunk
#define WTH      512              // threads per workgroup (16 wave32)
#define EPT      (CHUNK / WTH)    // 32 elements per thread (blocked)
#define ROWS     32               // kernel length
#define NPATH    30               // MAX_PATH
#define EPS_F    0.5f
#define INF_F    1e30f
#define CAND_CAP (1 << 16)
#define SENT     0xFFFFFFFFFFFFFFFFull

static_assert(CHUNK == WTH * EPT, "chunk/thread mismatch");

// min-plus carry map: f(D) = P + min(D, m), with start index s of the min.
struct Agg { float P; float m; int s; };

__device__ __forceinline__ Agg agg_combine(Agg L, Agg R) {
  // apply L (earlier) then R (later): P = PL+PR, m = min(mL, mR - PL), tie -> earlier (L)
  Agg o;
  o.P = L.P + R.P;
  float rm = R.m - L.P;
  bool keepL = (L.m <= rm);
  o.m = keepL ? L.m : rm;
  o.s = keepL ? L.s : R.s;
  return o;
}

__device__ __forceinline__ Agg agg_shfl_up(Agg a, int off) {
  Agg o;
  o.P = __shfl_up(a.P, off);
  o.m = __shfl_up(a.m, off);
  o.s = __shfl_up(a.s, off);
  return o;
}

// ---------------- init: carry flags + candidate counter ----------------
__global__ void spring_init_kernel(unsigned long long* __restrict__ carry, int m,
                                   int* __restrict__ cnt) {
  int i = blockIdx.x * blockDim.x + threadIdx.x;
  if (i == 0) *cnt = 0;
  for (; i < m; i += gridDim.x * blockDim.x) carry[i] = SENT;
}

// ---------------- fused wavefront DP ----------------
__global__ __launch_bounds__(WTH) void spring_dp_kernel(
    const float* __restrict__ x, const float* __restrict__ kern,
    float* __restrict__ Dbuf, int* __restrict__ Sbuf,
    unsigned long long* __restrict__ carry, int n) {
  extern __shared__ float xs[];          // CHUNK floats, layout xs[e*WTH + tid] (conflict-free)
  __shared__ float ksh[ROWS];
  __shared__ Agg   wagg[WTH / 32];
  __shared__ float sDc, sDg;
  __shared__ int   sSc, sSg;

  const int t    = blockIdx.x;           // chunk id
  const int tid  = threadIdx.x;
  const int lane = tid & 31;
  const int wid  = tid >> 5;
  const int cb   = t * CHUNK;            // chunk base (fits in int: n = 4M)
  const int j0   = cb + tid * EPT;       // this thread's first element

  if (tid < ROWS) ksh[tid] = kern[tid];

  // ---- stage x tile into LDS with CDNA5 async global->LDS copies (transposed) ----
  {
    const float* gp = x + j0;
    unsigned int l0 = (unsigned int)(size_t)(&xs[tid]);   // flat low 32 bits == LDS byte offset
#pragma unroll
    for (int rr = 0; rr < EPT; ++rr) {
      unsigned int loff = l0 + (unsigned int)(rr * WTH * 4);
      const float* ga = gp + rr;
      asm volatile("global_load_async_to_lds_b32 %0, %1, off"
                   :: "v"(loff), "v"(ga) : "memory");
    }
    asm volatile("s_wait_asynccnt 0" ::: "memory");
  }
  __syncthreads();

  // ---- row 0: D0 = (k0 - x)^2, S0 = j ----
  const float k0 = ksh[0];
#pragma unroll
  for (int e = 0; e < EPT; ++e) {
    int j = j0 + e;
    float xv = xs[e * WTH + tid];
    float dd = k0 - xv; dd *= dd;
    Dbuf[j] = dd;                        // parity 0
    Sbuf[j] = j;
  }
  // diag boundary for row 1 = D[0, cb-1], computable locally (no wait needed)
  float diagD = INF_F; int diagS = 0;    // meaningful in tid 0 only
  if (tid == 0 && t > 0) {
    float xm = x[cb - 1];
    float dd = k0 - xm; dd *= dd;
    diagD = dd; diagS = cb - 1;
  }
  __threadfence_block();
  __syncthreads();

  // ---- rows 1..31 ----
  for (int r = 1; r < ROWS; ++r) {
    const float ki = ksh[r];
    const float* __restrict__ Dp = Dbuf + (size_t)((r - 1) & 1) * n;
    const int*   __restrict__ Sp = Sbuf + (size_t)((r - 1) & 1) * n;
    float* __restrict__ Dq = Dbuf + (size_t)(r & 1) * n;
    int*   __restrict__ Sq = Sbuf + (size_t)(r & 1) * n;

    if (tid == 0) {
      float Dc = INF_F; int Sc = 0;
      if (t > 0) {
        const unsigned long long* cp = &carry[(size_t)(t - 1) * ROWS + r];
        unsigned long long v;
        for (;;) {
          v = __hip_atomic_load(cp, __ATOMIC_ACQUIRE, __HIP_MEMORY_SCOPE_AGENT);
          if (v != SENT) break;
          __builtin_amdgcn_s_sleep(1);
        }
        Dc = __uint_as_float((unsigned int)(v >> 32));
        Sc = (int)(unsigned int)v;
      }
      sDc = Dc; sSc = Sc; sDg = diagD; sSg = diagS;
      diagD = Dc; diagS = Sc;            // this row's in-carry is next row's diag boundary
    }
    __syncthreads();
    const float Dc  = sDc;  const int Sc  = sSc;
    const float dgD = sDg;  const int dgS = sSg;

    // ---- pass 1: c = min(up,diag) w/ start, local excl-cumsum, local min of (c - Pexcl) ----
    float cA[EPT]; int csA[EPT];
    float Pl = 0.f, mloc = INF_F; int sloc = 0;
    float Dm1; int Sm1;
    if (tid == 0) { Dm1 = dgD; Sm1 = dgS; }
    else          { Dm1 = Dp[j0 - 1]; Sm1 = Sp[j0 - 1]; }
#pragma unroll
    for (int e = 0; e < EPT; ++e) {
      int j = j0 + e;
      float xv = xs[e * WTH + tid];
      float dd = ki - xv; dd *= dd;
      float Dv = Dp[j]; int Sv = Sp[j];
      bool take = Dm1 < Dv;              // strict: tie keeps "up" (matches reference)
      float c  = take ? Dm1 : Dv;
      int   cs = take ? Sm1 : Sv;
      cA[e] = c; csA[e] = cs;
      float a = c - Pl;
      if (a < mloc) { mloc = a; sloc = cs; }   // strict: earliest index wins ties
      Pl += dd;
      Dm1 = Dv; Sm1 = Sv;
    }

    // ---- exclusive block scan of (P, m, s) over 512 threads (wave32 shfl + LDS) ----
    Agg a; a.P = Pl; a.m = mloc; a.s = sloc;
#pragma unroll
    for (int off = 1; off < 32; off <<= 1) {
      Agg o = agg_shfl_up(a, off);
      if (lane >= off) a = agg_combine(o, a);
    }
    Agg lex;                             // lane-exclusive within wave
    {
      Agg o = agg_shfl_up(a, 1);
      if (lane == 0) { lex.P = 0.f; lex.m = INF_F; lex.s = 0; } else lex = o;
    }
    if (lane == 31) wagg[wid] = a;
    __syncthreads();
    if (wid == 0) {
      Agg w;
      if (lane < (WTH / 32)) w = wagg[lane];
      else { w.P = 0.f; w.m = INF_F; w.s = 0; }
#pragma unroll
      for (int off = 1; off < (WTH / 32); off <<= 1) {
        Agg o = agg_shfl_up(w, off);
        if (lane >= off) w = agg_combine(o, w);
      }
      Agg o = agg_shfl_up(w, 1);
      Agg wex;
      if (lane == 0) { wex.P = 0.f; wex.m = INF_F; wex.s = 0; } else wex = o;
      if (lane < (WTH / 32)) wagg[lane] = wex;
    }
    __syncthreads();
    Agg ex = agg_combine(wagg[wid], lex);
    const float B = ex.P, Em = ex.m; const int Es = ex.s;

    // ---- pass 2: fold carry, emit D/S; tail thread publishes chunk carry early ----
    float Pl2 = 0.f, gm = INF_F; int gs = 0;
    float lastD = 0.f; int lastS = 0;
#pragma unroll
    for (int e = 0; e < EPT; ++e) {
      int j = j0 + e;
      float xv = xs[e * WTH + tid];
      float dd = ki - xv; dd *= dd;
      float aa = cA[e] - Pl2;
      if (aa < gm) { gm = aa; gs = csA[e]; }
      float gmg = gm - B;                // to global units
      float mv; int ms;
      if (Em <= gmg) { mv = Em; ms = Es; } else { mv = gmg; ms = gs; }
      float Pincl = B + Pl2 + dd;
      float Dn; int Sn;
      if (Dc < mv) { Dn = Pincl + Dc; Sn = Sc; }
      else         { Dn = Pincl + mv; Sn = ms; }
      Dq[j] = Dn; Sq[j] = Sn;
      Pl2 += dd;
      lastD = Dn; lastS = Sn;
    }
    if (tid == WTH - 1) {
      unsigned long long pk =
          ((unsigned long long)__float_as_uint(lastD) << 32) | (unsigned int)(unsigned)lastS;
      __hip_atomic_store(&carry[(size_t)t * ROWS + r], pk,
                         __ATOMIC_RELEASE, __HIP_MEMORY_SCOPE_AGENT);
    }
    __threadfence_block();
    __syncthreads();
  }
}

// ---------------- candidate compaction: D <= eps ----------------
__global__ void spring_cand_kernel(const float* __restrict__ Dfin,
                                   float* __restrict__ candD, int* __restrict__ candI,
                                   int* __restrict__ cnt, int n) {
  for (int j = blockIdx.x * blockDim.x + threadIdx.x; j < n;
       j += gridDim.x * blockDim.x) {
    float v = Dfin[j];
    if (v <= EPS_F) {
      int p = atomicAdd(cnt, 1);
      if (p < CAND_CAP) { candD[p] = v; candI[p] = j; }
    }
  }
}

// ---------------- 30-round stable argmin extraction ----------------
__global__ void spring_select_kernel(float* __restrict__ candD,
                                     const int* __restrict__ candI,
                                     const int* __restrict__ cnt,
                                     const int* __restrict__ Sfin,
                                     float* __restrict__ cost,
                                     int* __restrict__ startA, int* __restrict__ endA) {
  __shared__ float sv[256];
  __shared__ int   sj[256];
  __shared__ int   sp[256];
  const int tid = threadIdx.x;
  int K = *cnt; if (K > CAND_CAP) K = CAND_CAP;

  for (int k = 0; k < NPATH; ++k) {
    float bv = INF_F; int bj = 0x7fffffff; int bp = -1;
    for (int i = tid; i < K; i += 256) {
      float v = candD[i]; int j = candI[i];
      if (v < bv || (v == bv && j < bj)) { bv = v; bj = j; bp = i; }
    }
    sv[tid] = bv; sj[tid] = bj; sp[tid] = bp;
    __syncthreads();
    for (int s = 128; s > 0; s >>= 1) {
      if (tid < s) {
        float v = sv[tid + s]; int j = sj[tid + s];
        if (v < sv[tid] || (v == sv[tid] && j < sj[tid])) {
          sv[tid] = v; sj[tid] = j; sp[tid] = sp[tid + s];
        }
      }
      __syncthreads();
    }
    if (tid == 0) {
      if (sp[0] >= 0 && sv[0] <= EPS_F) {
        cost[k]   = sv[0];
        endA[k]   = sj[0];
        startA[k] = Sfin[sj[0]];
        candD[sp[0]] = INF_F;            // remove from pool
      } else {
        cost[k] = INF_F; endA[k] = 0; startA[k] = 0;
      }
    }
    __syncthreads();
  }
}

// ---------------- paint: smallest-cost covering interval wins ----------------
__global__ void spring_paint_kernel(float* __restrict__ out,
                                    const float* __restrict__ cost,
                                    const int* __restrict__ startA,
                                    const int* __restrict__ endA, int n) {
  __shared__ float c_[NPATH];
  __shared__ int   s_[NPATH], e_[NPATH];
  if (threadIdx.x < NPATH) {
    c_[threadIdx.x] = cost[threadIdx.x];
    s_[threadIdx.x] = startA[threadIdx.x];
    e_[threadIdx.x] = endA[threadIdx.x];
  }
  __syncthreads();
  int j = blockIdx.x * blockDim.x + threadIdx.x;
  if (j >= n) return;
  float o = 0.f;
  for (int k = 0; k < NPATH; ++k) {      // ascending cost: first hit = best = ref's overwrite order
    if (c_[k] <= EPS_F && j >= s_[k] && j < e_[k]) { o = c_[k]; break; }
  }
  out[j] = o;
}

// ---------------- launcher ----------------
extern "C" void kernel_launch(void* const* d_in, const int* in_sizes, int n_in,
                              void* d_out, int out_size, void* d_ws, size_t ws_size,
                              hipStream_t stream) {
  const float* x    = (const float*)d_in[0];
  const float* kern = (const float*)d_in[1];
  float* out = (float*)d_out;
  const int n  = in_sizes[0];
  const int NC = n / CHUNK;

  // workspace carve-up (ping-pong D/S stay L2-resident: 64 MB << 192 MB L2)
  char* p = (char*)d_ws;
  float* Dbuf = (float*)p;                p += (size_t)2 * n * sizeof(float);
  int*   Sbuf = (int*)p;                  p += (size_t)2 * n * sizeof(int);
  unsigned long long* carry = (unsigned long long*)p;
                                          p += (size_t)NC * ROWS * sizeof(unsigned long long);
  float* candD = (float*)p;               p += (size_t)CAND_CAP * sizeof(float);
  int*   candI = (int*)p;                 p += (size_t)CAND_CAP * sizeof(int);
  int*   cnt   = (int*)p;                 p += 256;
  float* cost  = (float*)p;               p += 32 * sizeof(float);
  int*   startA = (int*)p;                p += 32 * sizeof(int);
  int*   endA   = (int*)p;                p += 32 * sizeof(int);
  (void)ws_size; (void)n_in; (void)out_size;

  spring_init_kernel<<<32, 256, 0, stream>>>(carry, NC * ROWS, cnt);
  spring_dp_kernel<<<NC, WTH, CHUNK * sizeof(float), stream>>>(x, kern, Dbuf, Sbuf, carry, n);
  // final row index 31 -> parity 1
  spring_cand_kernel<<<512, 256, 0, stream>>>(Dbuf + (size_t)n, candD, candI, cnt, n);
  spring_select_kernel<<<1, 256, 0, stream>>>(candD, candI, cnt, Sbuf + (size_t)n,
                                              cost, startA, endA);
  spring_paint_kernel<<<(n + 255) / 256, 256, 0, stream>>>(out, cost, startA, endA, n);
}